// LocalCorrOptDiMP_85023172592492
// MI455X (gfx1250) — compile-verified
//
#include <hip/hip_runtime.h>

typedef __attribute__((ext_vector_type(2))) float v2f;
typedef __attribute__((ext_vector_type(8))) float v8f;

// Problem dims
// B=16, C=256, H=W=32, K=81 (9x9 window), PAD=4
static constexpr size_t NS = 16ull * 81 * 32 * 32;   // 1,327,104 score elements
static constexpr size_t NF = 16ull * 256 * 32 * 32;  // 4,194,304 filter elements

// Workspace layout (float offsets)
static constexpr size_t OFF_S     = 0;                     // scores / scores_grad [B,81,H,W]
static constexpr size_t OFF_M     = OFF_S + NS;            // mapped residuals     [B,81,H,W]
static constexpr size_t OFF_WSM   = OFF_M + NS;            // spatial_w*score_mask [B,81,H,W]
static constexpr size_t OFF_G     = OFF_WSM + NS;          // filter grad          [B,C,H,W]
static constexpr size_t OFF_ALPHA = OFF_G + NF;            // alpha                [B,H,W]
static constexpr size_t OFF_LBL   = OFF_ALPHA + 16384;     // label_map[81]
static constexpr size_t OFF_A1    = OFF_LBL + 81;          // (1-m)/2 [81]
static constexpr size_t OFF_A2    = OFF_A1 + 81;           // (1+m)/2 [81]
static constexpr size_t OFF_SW    = OFF_A2 + 81;           // spatial weight [81]
static constexpr size_t OFF_SCAL  = OFF_SW + 81;           // 0:step 1:reg_weight 2..4:src 5..7:regsum
static constexpr size_t OFF_P1    = OFF_SCAL + 8;          // 5184 block partials (src)
static constexpr size_t OFF_P2    = OFF_P1 + 5184;         // 512 block partials (flt^2)

// ---------------------------------------------------------------------------
// Prep: distance-map constants, step length, reg weight
// ---------------------------------------------------------------------------
__global__ void k_prep(const float* __restrict__ lw, const float* __restrict__ mw,
                       const float* __restrict__ sww, const float* __restrict__ lsl,
                       const float* __restrict__ fr, float* __restrict__ ws) {
  int t = threadIdx.x;
  if (t < 81) {
    int dy = t / 9, dx = t % 9;
    float dist = sqrtf((float)((dy - 4) * (dy - 4) + (dx - 4) * (dx - 4)));
    float r = 2.0f * dist;  // dist / BIN_DISP
    float lbl = 0.f, mv = 0.f, swv = 0.f;
    #pragma unroll
    for (int d = 0; d < 9; d++) {
      float bv = fmaxf(1.0f - fabsf(r - (float)d), 0.0f);
      lbl += bv * lw[d]; mv += bv * mw[d]; swv += bv * sww[d];
    }
    float bv9 = fminf(fmaxf(r - 8.0f, 0.0f), 1.0f);  // clip(1 + (r - 9), 0, 1)
    lbl += bv9 * lw[9]; mv += bv9 * mw[9]; swv += bv9 * sww[9];
    float m = 1.0f / (1.0f + expf(-mv));
    ws[OFF_LBL + t] = lbl;
    ws[OFF_A1 + t]  = (1.0f - m) * 0.5f;
    ws[OFF_A2 + t]  = (1.0f + m) * 0.5f;
    ws[OFF_SW + t]  = swv;
  } else if (t == 81) {
    ws[OFF_SCAL + 0] = expf(lsl[0]);
  } else if (t == 82) {
    float rr = fr[0] * fr[0];
    ws[OFF_SCAL + 1] = fmaxf(rr, 1e-10f) * (1.0f / 65536.0f);  // / feat_dim^2
  }
}

// ---------------------------------------------------------------------------
// Local correlation via f32 WMMA Gram tiles.
// One wave per (b, y, dy, m0): D[16x16] = A[16xK] x B[Kx16] accumulated K=256
// in 64 steps of V_WMMA_F32_16X16X4_F32. Two overlapping N-tiles cover the
// 9-wide diagonal band; lanes scatter valid (x, dx) entries to out.
// ---------------------------------------------------------------------------
__global__ __launch_bounds__(256) void k_corr(const float* __restrict__ F,
                                              const float* __restrict__ feat,
                                              float* __restrict__ out,
                                              const float* __restrict__ mult) {
  int lane = threadIdx.x & 31;
  int wid  = blockIdx.x * 8 + (threadIdx.x >> 5);   // 0..9215
  int t  = wid & 1;
  int q  = wid >> 1;        // (b*32 + y)*9 + dy
  int dy = q % 9;
  int by = q / 9;
  int y  = by & 31;
  int b  = by >> 5;
  int m0 = t << 4;
  int half = lane >> 4;     // K sub-group (K 0,1 vs 2,3)
  int lx   = lane & 15;
  int yrow = y + dy - 4;

  // B-tile columns (shifted x coordinate u), with zero-mask for x-padding
  int u0 = m0 - 4 + lx;
  int u1 = m0 + 4 + lx;
  float msk0 = (u0 >= 0 && u0 < 32) ? 1.0f : 0.0f;
  float msk1 = (u1 >= 0 && u1 < 32) ? 1.0f : 0.0f;
  int u0c = min(max(u0, 0), 31);
  int u1c = min(max(u1, 0), 31);

  v8f acc0 = {};
  v8f acc1 = {};
  if (yrow >= 0 && yrow < 32) {   // wave-uniform: OOB row -> all-zero scores
    const float* pA  = F    + (size_t)b * 262144 + (size_t)y * 32 + m0 + lx + (size_t)(half * 2) * 1024;
    const float* pB0 = feat + (size_t)b * 262144 + (size_t)yrow * 32 + u0c + (size_t)(half * 2) * 1024;
    const float* pB1 = feat + (size_t)b * 262144 + (size_t)yrow * 32 + u1c + (size_t)(half * 2) * 1024;
    for (int c0 = 0; c0 < 256; c0 += 4) {
      v2f a, b0, b1;
      a.x  = pA[0];          a.y  = pA[1024];
      b0.x = pB0[0] * msk0;  b0.y = pB0[1024] * msk0;
      b1.x = pB1[0] * msk1;  b1.y = pB1[1024] * msk1;
      pA += 4096; pB0 += 4096; pB1 += 4096;
      acc0 = __builtin_amdgcn_wmma_f32_16x16x4_f32(false, a, false, b0, (short)0, acc0, false, false);
      acc1 = __builtin_amdgcn_wmma_f32_16x16x4_f32(false, a, false, b1, (short)0, acc1, false, false);
    }
  }

  // Scatter band entries. C/D layout: lane holds col N=lx, rows M=half*8+v.
  int N  = lx;
  int Mb = half * 8;
  size_t obase = ((size_t)b * 81 + (size_t)dy * 9) * 1024 + (size_t)y * 32 + m0;
  #pragma unroll
  for (int v = 0; v < 8; v++) {
    int M = Mb + v;
    int dx0 = N - M;                       // tile0: u = x + dx - 4, col = M + dx
    if (dx0 >= 0 && dx0 <= 8) {
      size_t oi = obase + (size_t)dx0 * 1024 + M;
      float val = acc0[v];
      if (mult) val *= mult[oi];
      out[oi] = val;
    }
    int dx1 = N - M + 8;                   // tile1 owns M+dx in [16,23] (N >= 8)
    if (N >= 8 && dx1 >= 0 && dx1 <= 8) {
      size_t oi = obase + (size_t)dx1 * 1024 + M;
      float val = acc1[v];
      if (mult) val *= mult[oi];
      out[oi] = val;
    }
  }
}

// ---------------------------------------------------------------------------
// Elementwise on scores: activation, mask, residuals; per-block src partials
// ---------------------------------------------------------------------------
__global__ __launch_bounds__(256) void k_elem1(const float* __restrict__ S,
                                               float* __restrict__ Mr,
                                               float* __restrict__ WSM,
                                               const float* __restrict__ ws,
                                               float* __restrict__ part) {
  __shared__ float red[256];
  size_t idx = (size_t)blockIdx.x * 256 + threadIdx.x;
  float loc = 0.f;
  if (idx < NS) {
    int k = (int)((idx >> 10) % 81);
    float s   = S[idx];
    float a1  = ws[OFF_A1 + k], a2 = ws[OFF_A2 + k];
    float sw  = ws[OFF_SW + k], lbl = ws[OFF_LBL + k];
    float sact  = a1 * fabsf(s) + a2 * s;
    float sgn   = (s > 0.f) ? 1.f : ((s < 0.f) ? -1.f : 0.f);
    float smask = a1 * sgn + a2;
    float lr    = sw * (sact - lbl);
    Mr[idx]  = smask * sw * lr;
    WSM[idx] = sw * smask;
    loc = lr * lr;
  }
  red[threadIdx.x] = loc;
  __syncthreads();
  for (int s2 = 128; s2 > 0; s2 >>= 1) {
    if (threadIdx.x < s2) red[threadIdx.x] += red[threadIdx.x + s2];
    __syncthreads();
  }
  if (threadIdx.x == 0) part[blockIdx.x] = red[0];
}

// Deterministic fixed-order final reduction of block partials
__global__ void k_red(const float* __restrict__ part, int n, float* __restrict__ dst) {
  __shared__ float red[256];
  float s = 0.f;
  for (int i = threadIdx.x; i < n; i += 256) s += part[i];
  red[threadIdx.x] = s;
  __syncthreads();
  for (int k = 128; k > 0; k >>= 1) {
    if (threadIdx.x < k) red[threadIdx.x] += red[threadIdx.x + k];
    __syncthreads();
  }
  if (threadIdx.x == 0) *dst = red[0];
}

// ---------------------------------------------------------------------------
// Adjoint correlation + reg term; also accumulates sum(flt^2) per block
// One block per (b, y); residuals [81 x 32] staged in LDS.
// ---------------------------------------------------------------------------
__global__ __launch_bounds__(256) void k_corrT(const float* __restrict__ Mr,
                                               const float* __restrict__ feat,
                                               const float* __restrict__ flt,
                                               float* __restrict__ G,
                                               const float* __restrict__ ws,
                                               float* __restrict__ part) {
  __shared__ float rlds[81 * 32];
  __shared__ float red[256];
  int b = blockIdx.x >> 5;
  int y = blockIdx.x & 31;
  for (int i = threadIdx.x; i < 81 * 32; i += 256) {
    int k = i >> 5, x = i & 31;
    rlds[i] = Mr[((size_t)b * 81 + k) * 1024 + (size_t)y * 32 + x];
  }
  __syncthreads();
  float rw = ws[OFF_SCAL + 1];
  float fsum = 0.f;
  for (int p = threadIdx.x; p < 8192; p += 256) {
    int c = p >> 5, x = p & 31;
    size_t fi = ((size_t)b * 256 + c) * 1024 + (size_t)y * 32 + x;
    float fv = flt[fi];
    fsum += fv * fv;
    float g = rw * fv;
    for (int dy = 0; dy < 9; dy++) {
      int yrow = y + dy - 4;
      if (yrow < 0 || yrow > 31) continue;
      const float* fr = feat + ((size_t)b * 256 + c) * 1024 + (size_t)yrow * 32;
      #pragma unroll
      for (int dx = 0; dx < 9; dx++) {
        int xx = x + dx - 4;
        if (xx >= 0 && xx < 32)
          g += rlds[(dy * 9 + dx) * 32 + x] * fr[xx];
      }
    }
    G[fi] = g;
  }
  red[threadIdx.x] = fsum;
  __syncthreads();
  for (int k = 128; k > 0; k >>= 1) {
    if (threadIdx.x < k) red[threadIdx.x] += red[threadIdx.x + k];
    __syncthreads();
  }
  if (threadIdx.x == 0) part[blockIdx.x] = red[0];
}

// ---------------------------------------------------------------------------
// Per-pixel alpha = num / clip(den + rw*num, 1e-8)
// ---------------------------------------------------------------------------
__global__ __launch_bounds__(256) void k_alpha(const float* __restrict__ G,
                                               const float* __restrict__ SG,
                                               const float* __restrict__ ws,
                                               float* __restrict__ AL) {
  int pid = blockIdx.x * 256 + threadIdx.x;   // 0..16383 = (b, y, x)
  int b  = pid >> 10;
  int yx = pid & 1023;
  float an = 0.f;
  const float* pg = G + (size_t)b * 262144 + yx;
  for (int c = 0; c < 256; c++) { float v = pg[(size_t)c * 1024]; an += v * v; }
  float ad = 0.f;
  const float* ps = SG + (size_t)b * 82944 + yx;   // 81*1024
  for (int k = 0; k < 81; k++) { float v = ps[(size_t)k * 1024]; ad += v * v; }
  float rw = ws[OFF_SCAL + 1];
  ad = fmaxf(ad + rw * an, 1e-8f);
  AL[pid] = an / ad;
}

// flt -= step * alpha * filter_grad
__global__ __launch_bounds__(256) void k_update(float* __restrict__ flt,
                                                const float* __restrict__ G,
                                                const float* __restrict__ AL,
                                                const float* __restrict__ ws) {
  size_t i = (size_t)blockIdx.x * 256 + threadIdx.x;
  float step = ws[OFF_SCAL + 0];
  int b  = (int)(i >> 18);
  int yx = (int)(i & 1023);
  flt[i] -= step * AL[b * 1024 + yx] * G[i];
}

// Loss traces into d_out tail
__global__ void k_fin(const float* __restrict__ ws, float* __restrict__ out) {
  int i = threadIdx.x;
  if (i < 3) {
    float src = 0.5f * ws[OFF_SCAL + 2 + i] * (1.0f / 16.0f);
    float reg = 0.5f * ws[OFF_SCAL + 1] * ws[OFF_SCAL + 5 + i] * (1.0f / 16.0f);
    out[4194304 + i]     = src + reg;
    out[4194304 + 3 + i] = src;
    out[4194304 + 6 + i] = reg;
  }
}

extern "C" void kernel_launch(void* const* d_in, const int* in_sizes, int n_in,
                              void* d_out, int out_size, void* d_ws, size_t ws_size,
                              hipStream_t stream) {
  const float* flt_in = (const float*)d_in[0];
  const float* feat   = (const float*)d_in[1];
  const float* lw     = (const float*)d_in[2];
  const float* mw     = (const float*)d_in[3];
  const float* sww    = (const float*)d_in[4];
  const float* lsl    = (const float*)d_in[5];
  const float* fr     = (const float*)d_in[6];
  float* out = (float*)d_out;
  float* ws  = (float*)d_ws;

  float* S   = ws + OFF_S;
  float* Mr  = ws + OFF_M;
  float* WSM = ws + OFF_WSM;
  float* G   = ws + OFF_G;
  float* AL  = ws + OFF_ALPHA;
  float* P1  = ws + OFF_P1;
  float* P2  = ws + OFF_P2;
  float* flt = out;  // final filter lives in d_out; updated in place

  hipMemcpyAsync(flt, flt_in, NF * sizeof(float), hipMemcpyDeviceToDevice, stream);
  k_prep<<<1, 128, 0, stream>>>(lw, mw, sww, lsl, fr, ws);

  for (int it = 0; it < 3; it++) {
    k_corr<<<1152, 256, 0, stream>>>(flt, feat, S, nullptr);
    k_elem1<<<5184, 256, 0, stream>>>(S, Mr, WSM, ws, P1);
    k_red<<<1, 256, 0, stream>>>(P1, 5184, ws + OFF_SCAL + 2 + it);
    k_corrT<<<512, 256, 0, stream>>>(Mr, feat, flt, G, ws, P2);
    k_red<<<1, 256, 0, stream>>>(P2, 512, ws + OFF_SCAL + 5 + it);
    k_corr<<<1152, 256, 0, stream>>>(G, feat, S, WSM);   // scores_grad, pre-scaled by sw*mask
    k_alpha<<<64, 256, 0, stream>>>(G, S, ws, AL);
    k_update<<<16384, 256, 0, stream>>>(flt, G, AL, ws);
  }
  k_fin<<<1, 8, 0, stream>>>(ws, out);
}